// GATLayer_75788992905472
// MI455X (gfx1250) — compile-verified
//
#include <hip/hip_runtime.h>
#include <math.h>

// GATConv forward (PyG semantics) for gfx1250 / MI455X.
// Main GEMM xl = x @ W^T runs on the CDNA5 matrix pipe via
// V_WMMA_F32_16X16X4_F32 (full fp32, K=4 per step, 32 steps for K=128).

#define IN_C   128
#define OUT_C  64
#define HEADS  2
#define HC     (HEADS * OUT_C)   // 128
#define NEG_SLOPE 0.2f

typedef float v2f __attribute__((ext_vector_type(2)));
typedef float v8f __attribute__((ext_vector_type(8)));

// ---------------------------------------------------------------- utilities

__global__ void k_fill(float* __restrict__ p, int n, float v) {
  int i = blockIdx.x * blockDim.x + threadIdx.x;
  if (i < n) p[i] = v;
}

// float atomic max via monotone int encoding (no CAS loop):
// non-negative floats compare as signed ints; negative floats compare
// reversed as unsigned ints.
__device__ __forceinline__ void atomicMaxFloat(float* addr, float v) {
  if (v >= 0.0f) atomicMax((int*)addr, __float_as_int(v));
  else           atomicMin((unsigned int*)addr, __float_as_uint(v));
}

// ------------------------------------------------- degree / self-loop attrs

__global__ void k_deg_scatter(const int* __restrict__ dst,
                              const float* __restrict__ eattr,
                              float* __restrict__ deg,
                              float* __restrict__ loopsum, int E) {
  int e = blockIdx.x * blockDim.x + threadIdx.x;
  if (e >= E) return;
  int d = dst[e];
  atomicAdd(deg + d, 1.0f);
  atomicAdd(loopsum + 2 * d + 0, eattr[2 * e + 0]);
  atomicAdd(loopsum + 2 * d + 1, eattr[2 * e + 1]);
}

__global__ void k_loop_attr(const float* __restrict__ deg,
                            const float* __restrict__ loopsum,
                            float* __restrict__ loopat, int N) {
  int n = blockIdx.x * blockDim.x + threadIdx.x;
  if (n >= N) return;
  float dv = fmaxf(deg[n], 1.0f);
  loopat[2 * n + 0] = loopsum[2 * n + 0] / dv;
  loopat[2 * n + 1] = loopsum[2 * n + 1] / dv;
}

// --------------------------------------------------- xl = x @ W^T  (WMMA)

// Block = 256 threads = 8 waves. Block b computes rows [16b,16b+16); wave w
// computes output columns [16w,16w+16). K=128 -> 32 WMMA_F32_16X16X4 steps.
//
// f32 16x4 A layout (ISA 7.12.2): lanes 0-15 hold row M=lane, VGPR0/1 = K,K+1;
// lanes 16-31 hold K+2,K+3. B (4x16) mirrors this with N=lane&15 per column.
// C/D: VGPR r = row (r + 8*(lane>=16)), col = lane&15.
__global__ void __launch_bounds__(256)
k_gemm_xl(const float* __restrict__ x, const float* __restrict__ W,
          float* __restrict__ xl) {
  const int wave = threadIdx.x >> 5;
  const int lane = threadIdx.x & 31;
  const int half = lane >> 4;      // 0: K,K+1   1: K+2,K+3
  const int l15  = lane & 15;
  const int m0 = blockIdx.x << 4;
  const int n0 = wave << 4;

  const float* arow = x + (size_t)(m0 + l15) * IN_C + 2 * half;  // A row (M)
  const float* brow = W + (size_t)(n0 + l15) * IN_C + 2 * half;  // B col = W row (N)

  v8f c = {0.f, 0.f, 0.f, 0.f, 0.f, 0.f, 0.f, 0.f};
#pragma unroll 8
  for (int k = 0; k < IN_C; k += 4) {
    v2f a = *(const v2f*)(arow + k);
    v2f b = *(const v2f*)(brow + k);
    // 8 args: (neg_a, A, neg_b, B, c_mod, C, reuse_a, reuse_b)
    c = __builtin_amdgcn_wmma_f32_16x16x4_f32(false, a, false, b,
                                              (short)0, c, false, false);
  }

  float* drow = xl + (size_t)(m0 + 8 * half) * HC + n0 + l15;
#pragma unroll
  for (int r = 0; r < 8; ++r) drow[(size_t)r * HC] = c[r];
}

// ------------------------------------------- per-node attention projections

__global__ void k_node_attn(const float* __restrict__ xl,
                            const float* __restrict__ att_src,
                            const float* __restrict__ att_dst,
                            float* __restrict__ a_src,
                            float* __restrict__ a_dst, int N) {
  int t = blockIdx.x * blockDim.x + threadIdx.x;
  if (t >= N * HEADS) return;
  int n = t >> 1, h = t & 1;
  const float* xr = xl + (size_t)n * HC + h * OUT_C;
  const float* as = att_src + h * OUT_C;
  const float* ad = att_dst + h * OUT_C;
  float s0 = 0.f, s1 = 0.f;
#pragma unroll 8
  for (int c = 0; c < OUT_C; ++c) {
    float v = xr[c];
    s0 += v * as[c];
    s1 += v * ad[c];
  }
  a_src[t] = s0;
  a_dst[t] = s1;
}

// P[h][j] = sum_c W_edge[h*64+c][j] * att_edge[h*64+c]  (collapses the edge GEMM)
__global__ void k_edge_P(const float* __restrict__ W_edge,
                         const float* __restrict__ att_edge,
                         float* __restrict__ P) {
  int t = threadIdx.x;
  if (t >= HEADS * 2) return;
  int h = t >> 1, j = t & 1;
  float s = 0.f;
  for (int c = 0; c < OUT_C; ++c)
    s += W_edge[(h * OUT_C + c) * 2 + j] * att_edge[h * OUT_C + c];
  P[t] = s;
}

// ------------------------------------------ attention logits + segment max

__global__ void k_alpha1(const int* __restrict__ src, const int* __restrict__ dst,
                         const float* __restrict__ eattr,
                         const float* __restrict__ loopat,
                         const float* __restrict__ a_src,
                         const float* __restrict__ a_dst,
                         const float* __restrict__ P,
                         float* __restrict__ ealpha, float* __restrict__ amax,
                         int E, int N) {
  int t = blockIdx.x * blockDim.x + threadIdx.x;
  if (t >= (E + N) * HEADS) return;
  int row = t >> 1, h = t & 1;
  int s, d;
  float e0, e1;
  if (row < E) {
    s = src[row]; d = dst[row];
    e0 = eattr[2 * row]; e1 = eattr[2 * row + 1];
  } else {
    s = d = row - E;
    e0 = loopat[2 * s]; e1 = loopat[2 * s + 1];
  }
  float v = a_src[s * 2 + h] + a_dst[d * 2 + h] + e0 * P[2 * h] + e1 * P[2 * h + 1];
  v = (v > 0.0f) ? v : NEG_SLOPE * v;     // leaky_relu
  ealpha[t] = v;
  atomicMaxFloat(amax + d * 2 + h, v);
}

// ------------------------------------------------- exp + segment denominator

__global__ void k_alpha2(const int* __restrict__ dst,
                         const float* __restrict__ amax,
                         float* __restrict__ ealpha,
                         float* __restrict__ denom, int E, int N) {
  int t = blockIdx.x * blockDim.x + threadIdx.x;
  if (t >= (E + N) * HEADS) return;
  int row = t >> 1, h = t & 1;
  int d = (row < E) ? dst[row] : (row - E);
  float ev = expf(ealpha[t] - amax[d * 2 + h]);
  ealpha[t] = ev;
  atomicAdd(denom + d * 2 + h, ev);
}

// ----------------------- normalize, emit alpha, gather xl[src], scatter-add

// 16 lanes per row, float4 per lane -> each row's 512B xl gather is contiguous.
// Heads fused: out[d][c] += a0*xl[s][0][c] + a1*xl[s][1][c]; /HEADS at finalize.
__global__ void k_scatter(const int* __restrict__ src, const int* __restrict__ dst,
                          const float* __restrict__ xl,
                          const float* __restrict__ ealpha,
                          const float* __restrict__ denom,
                          float* __restrict__ out, float* __restrict__ alpha_out,
                          int E, int N) {
  int t = blockIdx.x * blockDim.x + threadIdx.x;
  if (t >= (E + N) * 16) return;
  int row = t >> 4, q = t & 15;
  int s, d;
  if (row < E) { s = src[row]; d = dst[row]; } else { s = d = row - E; }
  float a0 = ealpha[2 * row]     / denom[2 * d];
  float a1 = ealpha[2 * row + 1] / denom[2 * d + 1];
  if (q == 0) {
    alpha_out[2 * row]     = a0;
    alpha_out[2 * row + 1] = a1;
  }
  const float4 x0 = *(const float4*)(xl + (size_t)s * HC + 4 * q);          // head 0
  const float4 x1 = *(const float4*)(xl + (size_t)s * HC + OUT_C + 4 * q);  // head 1
  float* o = out + (size_t)d * OUT_C + 4 * q;
  atomicAdd(o + 0, a0 * x0.x + a1 * x1.x);
  atomicAdd(o + 1, a0 * x0.y + a1 * x1.y);
  atomicAdd(o + 2, a0 * x0.z + a1 * x1.z);
  atomicAdd(o + 3, a0 * x0.w + a1 * x1.w);
}

__global__ void k_finalize(float* __restrict__ out, const float* __restrict__ bias,
                           int N) {
  int t = blockIdx.x * blockDim.x + threadIdx.x;
  if (t >= N * OUT_C) return;
  out[t] = out[t] * (1.0f / HEADS) + bias[t & (OUT_C - 1)];
}

// ---------------------------------------------------------------- launcher

static inline int cdiv(long a, long b) { return (int)((a + b - 1) / b); }

extern "C" void kernel_launch(void* const* d_in, const int* in_sizes, int n_in,
                              void* d_out, int out_size, void* d_ws, size_t ws_size,
                              hipStream_t stream) {
  const float* x        = (const float*)d_in[0];
  const int*   eidx     = (const int*)d_in[1];
  const float* eattr    = (const float*)d_in[2];
  const float* W        = (const float*)d_in[3];
  const float* W_edge   = (const float*)d_in[4];
  const float* att_src  = (const float*)d_in[5];
  const float* att_dst  = (const float*)d_in[6];
  const float* att_edge = (const float*)d_in[7];
  const float* bias     = (const float*)d_in[8];

  const int N_ = in_sizes[0] / IN_C;   // 50000
  const int E_ = in_sizes[1] / 2;      // 800000
  const int* srcp = eidx;
  const int* dstp = eidx + E_;

  // workspace layout (floats)
  float* ws       = (float*)d_ws;
  float* xl       = ws;                                // N*128
  float* a_src_v  = xl + (size_t)N_ * HC;              // N*2
  float* a_dst_v  = a_src_v + (size_t)N_ * HEADS;      // N*2
  float* deg      = a_dst_v + (size_t)N_ * HEADS;      // N
  float* loopsum  = deg + N_;                          // N*2
  float* loopat   = loopsum + (size_t)N_ * 2;          // N*2
  float* ealpha   = loopat + (size_t)N_ * 2;           // (E+N)*2
  float* amax     = ealpha + (size_t)(E_ + N_) * 2;    // N*2
  float* denom    = amax + (size_t)N_ * 2;             // N*2
  float* P        = denom + (size_t)N_ * 2;            // 4

  float* out       = (float*)d_out;                    // N*64
  float* alpha_out = out + (size_t)N_ * OUT_C;         // (E+N)*2

  const int B = 256;

  // init accumulators (ws/d_out are poisoned by the harness)
  k_fill<<<cdiv(N_, B), B, 0, stream>>>(deg, N_, 0.f);
  k_fill<<<cdiv(2 * N_, B), B, 0, stream>>>(loopsum, 2 * N_, 0.f);
  k_fill<<<cdiv(2 * N_, B), B, 0, stream>>>(denom, 2 * N_, 0.f);
  k_fill<<<cdiv(2 * N_, B), B, 0, stream>>>(amax, 2 * N_, -3.0e38f);
  k_fill<<<cdiv(N_ * OUT_C, B), B, 0, stream>>>(out, N_ * OUT_C, 0.f);

  // self-loop attrs (fill_value='mean')
  k_deg_scatter<<<cdiv(E_, B), B, 0, stream>>>(dstp, eattr, deg, loopsum, E_);
  k_loop_attr<<<cdiv(N_, B), B, 0, stream>>>(deg, loopsum, loopat, N_);

  // main GEMM on the matrix pipe
  k_gemm_xl<<<N_ / 16, 256, 0, stream>>>(x, W, xl);

  // attention pieces
  k_node_attn<<<cdiv(N_ * HEADS, B), B, 0, stream>>>(xl, att_src, att_dst,
                                                     a_src_v, a_dst_v, N_);
  k_edge_P<<<1, 32, 0, stream>>>(W_edge, att_edge, P);
  k_alpha1<<<cdiv((long)(E_ + N_) * HEADS, B), B, 0, stream>>>(
      srcp, dstp, eattr, loopat, a_src_v, a_dst_v, P, ealpha, amax, E_, N_);
  k_alpha2<<<cdiv((long)(E_ + N_) * HEADS, B), B, 0, stream>>>(
      dstp, amax, ealpha, denom, E_, N_);

  // message passing (bandwidth-dominant phase)
  k_scatter<<<cdiv((long)(E_ + N_) * 16, B), B, 0, stream>>>(
      srcp, dstp, xl, ealpha, denom, out, alpha_out, E_, N_);

  // mean over heads + bias
  k_finalize<<<cdiv(N_ * OUT_C, B), B, 0, stream>>>(out, bias, N_);
}